// topic_model_toy_49314814493056
// MI455X (gfx1250) — compile-verified
//
#include <hip/hip_runtime.h>
#include <hip/hip_bf16.h>

typedef __attribute__((ext_vector_type(2))) float v2f;
typedef __attribute__((ext_vector_type(8))) float v8f;

#define THRES    0.3f
#define EPS_SUM  0.001f
#define EPS_NORM 1e-12f
#define Bn   64
#define Dn   1024
#define Sn   512
#define Kn   100
#define KP   112
#define Hn   2048
#define BS   32768   // B*S
#define NEG_INF (-3.402823466e38f)

// ---------------- tv_n: normalize topic_vector columns, pad K 100->112 with 0
__global__ __launch_bounds__(256)
void tvnorm_kernel(const float* __restrict__ tv, float* __restrict__ tvn) {
    int k = blockIdx.x;                 // 0..111
    int t = threadIdx.x;
    __shared__ float red[256];
    if (k >= Kn) {
        for (int d = t; d < Dn; d += 256) tvn[(size_t)d * KP + k] = 0.f;
        return;
    }
    float ss = 0.f;
    for (int d = t; d < Dn; d += 256) { float v = tv[(size_t)d * Kn + k]; ss += v * v; }
    red[t] = ss; __syncthreads();
    for (int o = 128; o > 0; o >>= 1) { if (t < o) red[t] += red[t + o]; __syncthreads(); }
    float rn = 1.f / fmaxf(sqrtf(red[0]), EPS_NORM);
    for (int d = t; d < Dn; d += 256) tvn[(size_t)d * KP + k] = tv[(size_t)d * Kn + k] * rn;
}

// ---------------- per (b,s) row inverse L2 norm of x
__global__ __launch_bounds__(256)
void rownorm_kernel(const float* __restrict__ f_input, float* __restrict__ invn) {
    int b = blockIdx.x >> 1;
    int s = ((blockIdx.x & 1) << 8) + threadIdx.x;
    const float* fb = f_input + (size_t)b * Dn * Sn;
    float ss = 0.f;
    for (int d = 0; d < Dn; ++d) { float v = fb[(size_t)d * Sn + s]; ss += v * v; }
    invn[b * Sn + s] = 1.f / fmaxf(sqrtf(ss), EPS_NORM);
}

// ---------------- concept_far = (sum(G) - K)/K^2, sum(G) = sum_d (rowsum_d)^2
__global__ __launch_bounds__(256)
void cfar_kernel(const float* __restrict__ tvn, float* __restrict__ out) {
    int t = threadIdx.x;
    __shared__ float red[256];
    float acc = 0.f;
    for (int d = t; d < Dn; d += 256) {
        float rs = 0.f;
        for (int k = 0; k < Kn; ++k) rs += tvn[(size_t)d * KP + k];
        acc += rs * rs;
    }
    red[t] = acc; __syncthreads();
    for (int o = 128; o > 0; o >>= 1) { if (t < o) red[t] += red[t + o]; __syncthreads(); }
    if (t == 0) out[130] = (red[0] - (float)Kn) / ((float)Kn * (float)Kn);
}

// ---------------- GEMM1: topic_prob = x @ tv_n via V_WMMA_F32_16X16X4_F32
// 128 threads (4 waves), M=64 rows (one b, 64 consecutive s), N=112.
// LDS tiles stored k-contiguous so each fragment is one aligned ds_load_b64.
// Register-staged double buffering over 32-wide k-chunks.
#define XS 34   // k-stride for Xs rows (even -> 8B aligned v2f loads)
#define WS 34
__global__ __launch_bounds__(128)
void gemm1_kernel(const float* __restrict__ f_input, const float* __restrict__ tvn,
                  const float* __restrict__ invn, float* __restrict__ tpn_ws,
                  float* __restrict__ out_tpnn) {
    __shared__ float Xs[2][64][XS];    // [buf][m][k]
    __shared__ float Ws[2][KP][WS];    // [buf][n][k]
    const int tid  = threadIdx.x;
    const int wave = tid >> 5;
    const int lane = tid & 31;
    const int l16  = lane & 15;
    const int half = lane >> 4;
    const int rowblk = blockIdx.x * 64;
    const int b  = rowblk >> 9;
    const int s0 = rowblk & (Sn - 1);
    const float* fb = f_input + (size_t)b * Dn * Sn;
    const int m0 = wave * 16;

    v8f acc[7] = {};
    float xr[16];
    float wr[28];

    // prologue: chunk 0 -> regs -> LDS buf 0
    #pragma unroll
    for (int j = 0; j < 16; ++j) {
        int i = tid + j * 128, kk = i >> 6, ss = i & 63;
        xr[j] = fb[(size_t)kk * Sn + s0 + ss];
    }
    #pragma unroll
    for (int j = 0; j < 28; ++j) {
        int i = tid + j * 128, kk = i / KP, n = i - kk * KP;
        wr[j] = tvn[(size_t)kk * KP + n];
    }
    #pragma unroll
    for (int j = 0; j < 16; ++j) {
        int i = tid + j * 128, kk = i >> 6, ss = i & 63;
        Xs[0][ss][kk] = xr[j];
    }
    #pragma unroll
    for (int j = 0; j < 28; ++j) {
        int i = tid + j * 128, kk = i / KP, n = i - kk * KP;
        Ws[0][n][kk] = wr[j];
    }
    __syncthreads();

    for (int c = 0; c < 32; ++c) {
        const int buf = c & 1;
        // issue global loads of chunk c+1 into registers (overlaps with WMMA)
        if (c < 31) {
            const int k0 = (c + 1) * 32;
            #pragma unroll
            for (int j = 0; j < 16; ++j) {
                int i = tid + j * 128, kk = i >> 6, ss = i & 63;
                xr[j] = fb[(size_t)(k0 + kk) * Sn + s0 + ss];
            }
            #pragma unroll
            for (int j = 0; j < 28; ++j) {
                int i = tid + j * 128, kk = i / KP, n = i - kk * KP;
                wr[j] = tvn[(size_t)(k0 + kk) * KP + n];
            }
        }
        // WMMA on buf
        #pragma unroll
        for (int ks = 0; ks < 32; ks += 4) {
            v2f a = *(const v2f*)&Xs[buf][m0 + l16][ks + 2 * half];
            #pragma unroll
            for (int tt = 0; tt < 7; ++tt) {
                v2f bb = *(const v2f*)&Ws[buf][tt * 16 + l16][ks + 2 * half];
                acc[tt] = __builtin_amdgcn_wmma_f32_16x16x4_f32(
                    false, a, false, bb, (short)0, acc[tt], false, false);
            }
        }
        // park chunk c+1 into the other buffer
        if (c < 31) {
            const int nb = (c + 1) & 1;
            #pragma unroll
            for (int j = 0; j < 16; ++j) {
                int i = tid + j * 128, kk = i >> 6, ss = i & 63;
                Xs[nb][ss][kk] = xr[j];
            }
            #pragma unroll
            for (int j = 0; j < 28; ++j) {
                int i = tid + j * 128, kk = i / KP, n = i - kk * KP;
                Ws[nb][n][kk] = wr[j];
            }
            __syncthreads();
        }
    }

    // epilogue: tpn = acc*invn, mask, row-normalize, emit tpn (ws) + tpnn (out)
    float inv[8];
    #pragma unroll
    for (int r = 0; r < 8; ++r) inv[r] = invn[rowblk + m0 + r + 8 * half];

    float am[7][8];
    float rs[8];
    #pragma unroll
    for (int r = 0; r < 8; ++r) rs[r] = 0.f;
    #pragma unroll
    for (int tt = 0; tt < 7; ++tt) {
        int n = tt * 16 + l16;
        #pragma unroll
        for (int r = 0; r < 8; ++r) {
            float tp  = acc[tt][r];
            float tpn = tp * inv[r];
            float v   = (tpn > THRES) ? tp : 0.f;
            am[tt][r] = v;
            rs[r] += v;
            if (n < Kn)
                tpn_ws[(size_t)(rowblk + m0 + r + 8 * half) * Kn + n] = tpn;
        }
    }
    #pragma unroll
    for (int r = 0; r < 8; ++r) {
        float v = rs[r];
        v += __shfl_xor(v, 1);
        v += __shfl_xor(v, 2);
        v += __shfl_xor(v, 4);
        v += __shfl_xor(v, 8);
        rs[r] = v + EPS_SUM;
    }
    #pragma unroll
    for (int tt = 0; tt < 7; ++tt) {
        int n = tt * 16 + l16;
        if (n < Kn) {
            #pragma unroll
            for (int r = 0; r < 8; ++r)
                out_tpnn[(size_t)(rowblk + m0 + r + 8 * half) * Kn + n] = am[tt][r] / rs[r];
        }
    }
}

// ---------------- GEMM2: sum_s relu(tpnn @ R1) -> per-rowblock partials (no atomics)
// 128 threads (4 waves), M=64 rows, N=64, full K=112 in LDS, k-contiguous layout.
#define K2S (KP + 2)   // even row stride -> aligned v2f loads
__global__ __launch_bounds__(128)
void gemm2_kernel(const float* __restrict__ tpnn, const float* __restrict__ R1,
                  float* __restrict__ featH_part) {
    __shared__ float As[64][K2S];    // [m][k]
    __shared__ float Bst[64][K2S];   // [n][k]
    __shared__ float csum[4][64];
    const int tid  = threadIdx.x;
    const int wave = tid >> 5;
    const int lane = tid & 31;
    const int l16  = lane & 15;
    const int half = lane >> 4;
    const int rowblk = blockIdx.x * 64;
    const int nblk   = blockIdx.y * 64;
    const int m0 = wave * 16;

    for (int i = tid; i < 64 * KP; i += 128) {
        int m = i / KP, k = i - m * KP;
        As[m][k] = (k < Kn) ? tpnn[(size_t)(rowblk + m) * Kn + k] : 0.f;
    }
    for (int i = tid; i < KP * 64; i += 128) {
        int k = i >> 6, n = i & 63;
        Bst[n][k] = (k < Kn) ? R1[(size_t)k * Hn + nblk + n] : 0.f;
    }
    __syncthreads();

    v8f acc[4] = {};
    #pragma unroll
    for (int ks = 0; ks < KP; ks += 4) {
        v2f a = *(const v2f*)&As[m0 + l16][ks + 2 * half];
        #pragma unroll
        for (int tt = 0; tt < 4; ++tt) {
            v2f bb = *(const v2f*)&Bst[tt * 16 + l16][ks + 2 * half];
            acc[tt] = __builtin_amdgcn_wmma_f32_16x16x4_f32(
                false, a, false, bb, (short)0, acc[tt], false, false);
        }
    }

    // relu + sum over this wave's 16 rows, then deterministic cross-wave reduce
    #pragma unroll
    for (int tt = 0; tt < 4; ++tt) {
        float cs = 0.f;
        #pragma unroll
        for (int r = 0; r < 8; ++r) cs += fmaxf(acc[tt][r], 0.f);
        cs += __shfl_xor(cs, 16);
        if (half == 0) csum[wave][tt * 16 + l16] = cs;
    }
    __syncthreads();
    if (tid < 64) {
        float v = csum[0][tid] + csum[1][tid] + csum[2][tid] + csum[3][tid];
        featH_part[(size_t)blockIdx.x * Hn + nblk + tid] = v;
    }
}

// ---------------- feat[b][d] = (1/S) * sum_h featH[b][h] * R2[h][d]
__global__ __launch_bounds__(256)
void feat_kernel(const float* __restrict__ featH_part, const float* __restrict__ R2,
                 float* __restrict__ feat) {
    int b = blockIdx.x;
    int d = blockIdx.y * 256 + threadIdx.x;
    const float* fp = featH_part + (size_t)b * 8 * Hn;   // 8 row-blocks per b
    float acc = 0.f;
    for (int h = 0; h < Hn; ++h) {
        float fh = fp[h] + fp[Hn + h] + fp[2 * Hn + h] + fp[3 * Hn + h]
                 + fp[4 * Hn + h] + fp[5 * Hn + h] + fp[6 * Hn + h] + fp[7 * Hn + h];
        acc += fh * R2[(size_t)h * Dn + d];
    }
    feat[(size_t)b * Dn + d] = acc * (1.f / (float)Sn);
}

// ---------------- pred = feat @ W_cls + b_cls ; also out[128] = 0
__global__ __launch_bounds__(256)
void pred_kernel(const float* __restrict__ feat, const float* __restrict__ W,
                 const float* __restrict__ bc, float* __restrict__ out) {
    int b = blockIdx.x, t = threadIdx.x;
    __shared__ float r0[256], r1[256];
    float a0 = 0.f, a1 = 0.f;
    for (int d = t; d < Dn; d += 256) {
        float f = feat[(size_t)b * Dn + d];
        a0 += f * W[d * 2 + 0];
        a1 += f * W[d * 2 + 1];
    }
    r0[t] = a0; r1[t] = a1; __syncthreads();
    for (int o = 128; o > 0; o >>= 1) {
        if (t < o) { r0[t] += r0[t + o]; r1[t] += r1[t + o]; }
        __syncthreads();
    }
    if (t == 0) {
        out[b * 2 + 0] = r0[0] + bc[0];
        out[b * 2 + 1] = r1[0] + bc[1];
        if (b == 0) out[128] = 0.0f;
    }
}

// ---------------- per-concept top-16 over 32768 values of topic_prob_n
__global__ __launch_bounds__(256)
void topk_kernel(const float* __restrict__ tpn, float* __restrict__ partial) {
    int k = blockIdx.x, t = threadIdx.x;
    float loc[16];
    #pragma unroll
    for (int i = 0; i < 16; ++i) loc[i] = NEG_INF;
    for (int r = t; r < BS; r += 256) {
        float v = tpn[(size_t)r * Kn + k];
        if (v > loc[15]) {
            loc[15] = v;
            #pragma unroll
            for (int i = 15; i > 0; --i)
                if (loc[i] > loc[i - 1]) { float tmp = loc[i]; loc[i] = loc[i - 1]; loc[i - 1] = tmp; }
        }
    }
    __shared__ float cand[256 * 16];
    __shared__ int   ptr[256];
    __shared__ float rv[256];
    __shared__ int   ri[256];
    #pragma unroll
    for (int i = 0; i < 16; ++i) cand[t * 16 + i] = loc[i];
    ptr[t] = 0;
    __syncthreads();
    float total = 0.f;
    for (int it = 0; it < 16; ++it) {
        float mv = (ptr[t] < 16) ? cand[t * 16 + ptr[t]] : NEG_INF;
        rv[t] = mv; ri[t] = t; __syncthreads();
        for (int o = 128; o > 0; o >>= 1) {
            if (t < o && rv[t + o] > rv[t]) { rv[t] = rv[t + o]; ri[t] = ri[t + o]; }
            __syncthreads();
        }
        if (t == 0) total += rv[0];
        if (t == ri[0]) ptr[t]++;
        __syncthreads();
    }
    if (t == 0) partial[k] = total;
}

// ---------------- concept_sim = -(sum of 100 per-concept top16 sums)/1600
__global__ __launch_bounds__(128)
void csim_kernel(const float* __restrict__ partial, float* __restrict__ out) {
    int t = threadIdx.x;
    __shared__ float red[128];
    float a = 0.f;
    for (int i = t; i < Kn; i += 128) a += partial[i];
    red[t] = a; __syncthreads();
    for (int o = 64; o > 0; o >>= 1) { if (t < o) red[t] += red[t + o]; __syncthreads(); }
    if (t == 0) out[129] = -(red[0] / ((float)Kn * (float)(Bn / 4)));
}

extern "C" void kernel_launch(void* const* d_in, const int* in_sizes, int n_in,
                              void* d_out, int out_size, void* d_ws, size_t ws_size,
                              hipStream_t stream) {
    const float* f_input = (const float*)d_in[0];
    const float* topic_v = (const float*)d_in[1];
    const float* R1      = (const float*)d_in[2];
    const float* R2      = (const float*)d_in[3];
    const float* W_cls   = (const float*)d_in[4];
    const float* b_cls   = (const float*)d_in[5];
    float* out = (float*)d_out;

    // workspace layout (floats)
    float* ws    = (float*)d_ws;
    float* tvn   = ws;                         // Dn*KP          = 114688
    float* invn  = tvn   + (size_t)Dn * KP;    // BS             = 32768
    float* tpn   = invn  + BS;                 // BS*Kn          = 3276800
    float* fHp   = tpn   + (size_t)BS * Kn;    // 512*Hn partial = 1048576
    float* feat  = fHp   + (size_t)512 * Hn;   // Bn*Dn          = 65536
    float* tkp   = feat  + (size_t)Bn * Dn;    // Kn             = 100

    float* out_tpnn = out + 131;               // pred[128], 0, csim, cfar, then tpnn

    tvnorm_kernel <<<KP, 256, 0, stream>>>(topic_v, tvn);
    rownorm_kernel<<<128, 256, 0, stream>>>(f_input, invn);
    cfar_kernel   <<<1, 256, 0, stream>>>(tvn, out);
    gemm1_kernel  <<<BS / 64, 128, 0, stream>>>(f_input, tvn, invn, tpn, out_tpnn);
    topk_kernel   <<<Kn, 256, 0, stream>>>(tpn, tkp);
    csim_kernel   <<<1, 128, 0, stream>>>(tkp, out);
    gemm2_kernel  <<<dim3(BS / 64, Hn / 64), 128, 0, stream>>>(out_tpnn, R1, fHp);
    feat_kernel   <<<dim3(Bn, Dn / 256), 256, 0, stream>>>(fHp, R2, feat);
    pred_kernel   <<<Bn, 256, 0, stream>>>(feat, W_cls, b_cls, out);
}